// TensorMessagePassingLayer_8400956030998
// MI455X (gfx1250) — compile-verified
//
#include <hip/hip_runtime.h>

#define N_NODES 100000
#define N_EDGES 1600000
#define DIM 64
#define EPB 16  // edges per 256-thread block (16 lanes per edge)

// ------------------------------------------------------------------
// Kernel 1: zero the accumulator (d_out) and the per-node counts.
// ------------------------------------------------------------------
__global__ __launch_bounds__(256) void mp_init(float4* __restrict__ out4,
                                               float* __restrict__ counts) {
    const int total4 = N_NODES * (DIM / 4);  // 1,600,000 float4
    int i = blockIdx.x * blockDim.x + threadIdx.x;
    if (i < total4) out4[i] = make_float4(0.f, 0.f, 0.f, 0.f);
    if (i < N_NODES) counts[i] = 0.f;
}

// ------------------------------------------------------------------
// Kernel 2: per-edge gather(src) + atomic scatter(dst).
//   Uses the identity  agg[v] = sum(f[src]) + deg[v]*f[v],  so only
//   the SRC row is gathered here; the deg*f term is folded into the
//   finalize pass. Halves the random-row read traffic.
//   - 256 threads = 16 half-waves; each half-wave owns one edge.
//   - Block's 16 src + 16 dst int64 indices (2 x 128B, contiguous)
//     are staged into LDS via gfx1250 async DMA (ASYNCcnt path).
//   - Lane l16 loads float4 [4*l16 .. 4*l16+3] of the src row (one
//     coalesced 256B transaction) and emits 4 native f32 atomic adds
//     into the dst accumulator row.
// ------------------------------------------------------------------
__global__ __launch_bounds__(256) void mp_scatter(const float* __restrict__ feat,
                                                  const long long* __restrict__ ei,
                                                  float* __restrict__ out,
                                                  float* __restrict__ counts) {
    __shared__ __align__(16) long long s_idx[2 * EPB];  // [0..15]=src, [16..31]=dst
    const int tid = threadIdx.x;
    const long long e0 = (long long)blockIdx.x * EPB;

    // ---- async-DMA stage of the index tile into LDS (lanes 0..15, b128 each)
    if (tid < 16) {
        const char* g = (tid < 8)
            ? (const char*)(ei + e0) + tid * 16                      // src row slice
            : (const char*)(ei + N_EDGES + e0) + (tid - 8) * 16;     // dst row slice
        unsigned lds = (unsigned)(unsigned long long)(&s_idx[tid * 2]);
        asm volatile("global_load_async_to_lds_b128 %0, %1, off"
                     :: "v"(lds), "v"((unsigned long long)g)
                     : "memory");
    }
    // ---- stream-prefetch index lines ~512 edges (4KB/row) ahead into L2
    if (tid == 16 && e0 + 512 < N_EDGES)
        __builtin_prefetch((const char*)(ei + e0 + 512), 0, 1);
    if (tid == 17 && e0 + 512 < N_EDGES)
        __builtin_prefetch((const char*)(ei + N_EDGES + e0 + 512), 0, 1);

    asm volatile("s_wait_asynccnt 0" ::: "memory");  // no-op in waves that issued nothing
    __syncthreads();

    const int slot = tid >> 4;  // edge slot within block (0..15)
    const int l16  = tid & 15;  // feature-quarter lane (0..15)
    const long long e = e0 + slot;
    if (e < N_EDGES) {
        const long long s = s_idx[slot];
        const long long d = s_idx[EPB + slot];
        const float4 a = *(const float4*)(feat + s * DIM + l16 * 4);
        float* o = out + d * DIM + l16 * 4;
        unsafeAtomicAdd(o + 0, a.x);   // global_atomic_add_f32 (no return)
        unsafeAtomicAdd(o + 1, a.y);
        unsafeAtomicAdd(o + 2, a.z);
        unsafeAtomicAdd(o + 3, a.w);
        if (l16 == 0) unsafeAtomicAdd(counts + d, 1.0f);
    }
}

// ------------------------------------------------------------------
// Kernel 3: out = f + (acc + cnt*f) / max(cnt, 1)   (float4 traffic)
//   (cnt==0  ->  acc==0  ->  out = f, matching the reference clamp)
// ------------------------------------------------------------------
__global__ __launch_bounds__(256) void mp_finalize(const float4* __restrict__ feat4,
                                                   const float* __restrict__ counts,
                                                   float4* __restrict__ out4) {
    const int total4 = N_NODES * (DIM / 4);
    int i = blockIdx.x * blockDim.x + threadIdx.x;
    if (i >= total4) return;
    const int node = i >> 4;  // 16 float4 per node
    const float cnt = counts[node];
    const float inv = 1.0f / fmaxf(cnt, 1.0f);
    const float4 acc = out4[i];
    const float4 f   = feat4[i];
    out4[i] = make_float4(fmaf(fmaf(cnt, f.x, acc.x), inv, f.x),
                          fmaf(fmaf(cnt, f.y, acc.y), inv, f.y),
                          fmaf(fmaf(cnt, f.z, acc.z), inv, f.z),
                          fmaf(fmaf(cnt, f.w, acc.w), inv, f.w));
}

extern "C" void kernel_launch(void* const* d_in, const int* in_sizes, int n_in,
                              void* d_out, int out_size, void* d_ws, size_t ws_size,
                              hipStream_t stream) {
    const float*     feat = (const float*)d_in[0];      // [N_NODES, 64] f32
    const long long* ei   = (const long long*)d_in[1];  // [2, N_EDGES] i64
    float* out    = (float*)d_out;                      // [N_NODES, 64] f32
    float* counts = (float*)d_ws;                       // N_NODES f32 scratch

    const int total4 = N_NODES * (DIM / 4);             // 1,600,000
    mp_init    <<<(total4 + 255) / 256, 256, 0, stream>>>((float4*)out, counts);
    mp_scatter <<<N_EDGES / EPB,        256, 0, stream>>>(feat, ei, out, counts);
    mp_finalize<<<(total4 + 255) / 256, 256, 0, stream>>>((const float4*)feat, counts,
                                                          (float4*)out);
}